// EpipolarSampler_5669356831211
// MI455X (gfx1250) — compile-verified
//
#include <hip/hip_runtime.h>
#include <math.h>

// ---------------------------------------------------------------------------
// Epipolar sampler for MI455X (gfx1250, wave32).
// Memory-bound: ~294 MB of outputs => ~13us floor at 23.3 TB/s HBM.
// No matmul structure -> no WMMA; the CDNA5 feature used is the async
// global->LDS DMA path (ASYNCcnt) to turn the NCHW channel-strided bilinear
// gather into conflict-free LDS reads.
// ---------------------------------------------------------------------------

#define BB 2
#define VV 2
#define CCH 128
#define HH 64
#define WW 64
#define RR (HH * WW)     // 4096 rays per view
#define SS 32            // samples per ray
#define EPSF 1e-6f

#define CHG 4            // channels staged per block (64 KB LDS slab)
#define RPB 256          // rays per sampler block

// ---- output layout (flat, reference tuple order) --------------------------
constexpr long long FEAT_N   = (long long)BB * VV * RR * SS * CCH; // 67,108,864
constexpr long long OVL_OFF  = FEAT_N;
constexpr long long OVL_N    = (long long)BB * VV * RR;
constexpr long long XYRAY_OFF = OVL_OFF + OVL_N;
constexpr long long XYRAY_N  = (long long)BB * VV * RR * 2;
constexpr long long XYS_OFF  = XYRAY_OFF + XYRAY_N;
constexpr long long XYS_N    = (long long)BB * VV * RR * SS * 2;
constexpr long long XYSN_OFF = XYS_OFF + XYS_N;
constexpr long long XYSF_OFF = XYSN_OFF + XYS_N;
constexpr long long ORG_OFF  = XYSF_OFF + XYS_N;
constexpr long long ORG_N    = (long long)BB * VV * RR * 3;
constexpr long long DIR_OFF  = ORG_OFF + ORG_N;

// ---- CDNA5 async global->LDS path -----------------------------------------
#if defined(__gfx1250__) && __has_builtin(__builtin_amdgcn_global_load_async_to_lds_b32)
#define USE_ASYNC_LDS 1
typedef __attribute__((address_space(1))) int as1_int;  // prints as "__device__ int *"
typedef __attribute__((address_space(3))) int as3_int;  // prints as "__shared__ int *"
#else
#define USE_ASYNC_LDS 0
#endif

__device__ __forceinline__ void wait_async_zero() {
#if USE_ASYNC_LDS
#if __has_builtin(__builtin_amdgcn_s_wait_asynccnt)
  __builtin_amdgcn_s_wait_asynccnt(0);
#else
  asm volatile("s_wait_asynccnt 0" ::: "memory");
#endif
#endif
}

__device__ __forceinline__ float fixnum_big(float v) {
  // jnp.nan_to_num(p, nan=0, posinf=1e8, neginf=-1e8)
  if (!(v == v)) return 0.0f;
  if (v == __builtin_inff()) return 1e8f;
  if (v == -__builtin_inff()) return -1e8f;
  return v;
}

__device__ __forceinline__ float fixnum_zero(float v) {
  // jnp.nan_to_num(xy, nan=0, posinf=0, neginf=0)
  if (!(v == v) || v == __builtin_inff() || v == -__builtin_inff()) return 0.0f;
  return v;
}

// ---------------------------------------------------------------------------
// Kernel 1: per-ray geometry. One thread per (b, v, ray).
// ---------------------------------------------------------------------------
__global__ __launch_bounds__(256) void ray_setup_kernel(
    const float* __restrict__ ex,    // (B,V,4,4)
    const float* __restrict__ Kin,   // (B,V,3,3)
    const float* __restrict__ nearp, // (B,V)
    const float* __restrict__ farp,  // (B,V)
    float* __restrict__ out)
{
  int gid = blockIdx.x * 256 + threadIdx.x;
  if (gid >= BB * VV * RR) return;
  int r  = gid & (RR - 1);
  int bv = gid >> 12;           // RR == 4096
  int vi = bv & (VV - 1);
  int bi = bv / VV;

  int px = r & (WW - 1), py = r >> 6;
  float xs = (px + 0.5f) / (float)WW;
  float ys = (py + 0.5f) / (float)HH;

  const float* Es = ex + (long long)bv * 16;
  const float* Ks = Kin + (long long)bv * 9;

  // d_cam = normalize(K^-1 @ [x,y,1])   (K is [[fx,0,cx],[0,fy,cy],[0,0,1]])
  float dx = (xs - Ks[2]) / Ks[0];
  float dy = (ys - Ks[5]) / Ks[4];
  float inv = 1.0f / sqrtf(dx * dx + dy * dy + 1.0f);
  dx *= inv; dy *= inv; float dz = inv;

  // world-space direction / origin
  float Dx = Es[0] * dx + Es[1] * dy + Es[2]  * dz;
  float Dy = Es[4] * dx + Es[5] * dy + Es[6]  * dz;
  float Dz = Es[8] * dx + Es[9] * dy + Es[10] * dz;
  float Ox = Es[3], Oy = Es[7], Oz = Es[11];

  // other view (v==2 -> index_v is simply 1-vi)
  int ov = (VV - 1) - vi;
  const float* Eo = ex + ((long long)bi * VV + ov) * 16;
  const float* Ko = Kin + ((long long)bi * VV + ov) * 9;
  float Kfx = Ko[0], Kcx = Ko[2], Kfy = Ko[4], Kcy = Ko[5];

  // w2c = inv(ext_o) = [R^T | -R^T t]  (ext_o is a rigid transform)
  float tx = Ox - Eo[3], ty = Oy - Eo[7], tz = Oz - Eo[11];
  float ocx = Eo[0] * tx + Eo[4] * ty + Eo[8]  * tz;
  float ocy = Eo[1] * tx + Eo[5] * ty + Eo[9]  * tz;
  float ocz = Eo[2] * tx + Eo[6] * ty + Eo[10] * tz;
  float dcx = Eo[0] * Dx + Eo[4] * Dy + Eo[8]  * Dz;
  float dcy = Eo[1] * Dx + Eo[5] * Dy + Eo[9]  * Dz;
  float dcz = Eo[2] * Dx + Eo[6] * Dy + Eo[10] * Dz;

  // four image-border intersections: (dim,cv) = (0,0),(0,1),(1,0),(1,1)
  float ts[4], xv[4], yv[4]; bool vd[4];
#pragma unroll
  for (int i = 0; i < 4; ++i) {
    int dim = i >> 1;
    float cv = (float)(i & 1);
    float fs, fo, cs, co, os_, oo, ds_, do_;
    if (dim == 0) { fs = Kfx; fo = Kfy; cs = Kcx; co = Kcy; os_ = ocx; oo = ocy; ds_ = dcx; do_ = dcy; }
    else          { fs = Kfy; fo = Kfx; cs = Kcy; co = Kcx; os_ = ocy; oo = ocx; ds_ = dcy; do_ = dcx; }
    float c = (cv - cs) / fs;
    float t = (c * ocz - os_) / (ds_ - c * dcz);
    float den = ds_ * ocz - dcz * os_;
    den = (den == 0.0f) ? EPSF : den;
    float other = co + fo * (oo * (ds_ - c * dcz) + do_ * (c * ocz - os_)) / den;
    ts[i] = t;
    xv[i] = (dim == 0) ? cv : other;
    yv[i] = (dim == 0) ? other : cv;
    vd[i] = (ocz + t * dcz > EPSF) && (t > EPSF);
  }

  // reduce: first-index argmin / argmax (invalid -> +/-inf, NaN never wins)
  float bmin = __builtin_inff(); int imin = 0;
  float bmax = -__builtin_inff(); int imax = 0;
#pragma unroll
  for (int i = 0; i < 4; ++i) {
    float tmn = vd[i] ? ts[i] : __builtin_inff();
    float tmx = vd[i] ? ts[i] : -__builtin_inff();
    if (tmn < bmin) { bmin = tmn; imin = i; }
    if (tmx > bmax) { bmax = tmx; imax = i; }
  }
  float fmin_x = xv[imin], fmin_y = yv[imin]; bool fmin_ok = vd[imin];
  float fmax_x = xv[imax], fmax_y = yv[imax]; bool fmax_ok = vd[imax];

  // near/far point projections
  float nearv = nearp[bv], farv = farp[bv];
  float nX, nY, fX, fY; bool nok, fok;
#pragma unroll
  for (int q = 0; q < 2; ++q) {
    float tv = q ? farv : nearv;
    float ptx = ocx + tv * dcx, pty = ocy + tv * dcy, ptz = ocz + tv * dcz;
    float den = ptz + 1e-8f;
    float px_ = fixnum_big(ptx / den);
    float py_ = fixnum_big(pty / den);
    float pz_ = fixnum_big(ptz / den);
    float X = Kfx * px_ + Kcx * pz_;
    float Y = Kfy * py_ + Kcy * pz_;
    bool ok = (ptz > EPSF) && (tv > EPSF);
    if (q) { fX = X; fY = Y; fok = ok; } else { nX = X; nY = Y; nok = ok; }
  }

  float xminx = nok ? nX : fmin_x, xminy = nok ? nY : fmin_y;
  float xmaxx = fok ? fX : fmax_x, xmaxy = fok ? fY : fmax_y;
  bool ovl = (nok || fmin_ok) && (fok || fmax_ok);
  float ovf = ovl ? 1.0f : 0.0f;
  xminx = fixnum_zero(xminx) * ovf; xminy = fixnum_zero(xminy) * ovf;
  xmaxx = fixnum_zero(xmaxx) * ovf; xmaxy = fixnum_zero(xmaxy) * ovf;
  float dXx = xmaxx - xminx, dXy = xmaxy - xminy;

  long long base = (long long)bv * RR + r;
  out[XYRAY_OFF + base * 2 + 0] = xs;
  out[XYRAY_OFF + base * 2 + 1] = ys;
  out[OVL_OFF + base] = ovf;
  out[ORG_OFF + base * 3 + 0] = Ox;
  out[ORG_OFF + base * 3 + 1] = Oy;
  out[ORG_OFF + base * 3 + 2] = Oz;
  out[DIR_OFF + base * 3 + 0] = Dx;
  out[DIR_OFF + base * 3 + 1] = Dy;
  out[DIR_OFF + base * 3 + 2] = Dz;

  const float half = 0.5f / (float)SS;
  long long sb = base * SS * 2;
#pragma unroll 4
  for (int k = 0; k < SS; ++k) {
    float dep = (k + 0.5f) / (float)SS;
    out[XYS_OFF  + sb + 2 * k + 0] = xminx + dep * dXx;
    out[XYS_OFF  + sb + 2 * k + 1] = xminy + dep * dXy;
    out[XYSN_OFF + sb + 2 * k + 0] = xminx + (dep - half) * dXx;
    out[XYSN_OFF + sb + 2 * k + 1] = xminy + (dep - half) * dXy;
    out[XYSF_OFF + sb + 2 * k + 0] = xminx + (dep + half) * dXx;
    out[XYSF_OFF + sb + 2 * k + 1] = xminy + (dep + half) * dXy;
  }
}

// ---------------------------------------------------------------------------
// Kernel 2: bilinear sampler. Block = (bv, ray tile of 256, 4-channel group).
// Stage 4 channels of the source image (64 KB) into LDS, channel-interleaved,
// via CDNA5 async global->LDS DMA; then 4-tap bilinear from LDS.
// feat[b,vi,0,ray,ss,c] = bilinear(image[b,1-vi,c], xy_sample[b,vi,0,ray,ss]) * ov
// ---------------------------------------------------------------------------
__global__ __launch_bounds__(256) void sampler_kernel(
    const float* __restrict__ images, // (B,V,C,H,W)
    float* __restrict__ out)
{
  __shared__ float tile[HH * WW * CHG]; // 65536 bytes

  int blk = blockIdx.x;
  int cg = blk & (CCH / CHG - 1);            // 0..31
  int rt = (blk >> 5) & (RR / RPB - 1);      // 0..15
  int bv = blk >> 9;                         // 0..3
  int vi = bv & (VV - 1);
  int bi = bv / VV;
  int src_bv = bi * VV + ((VV - 1) - vi);    // sample the OTHER view's image

  const float* img = images + ((long long)src_bv * CCH + (long long)cg * CHG) * (HH * WW);

  int tid = threadIdx.x;
  // stage slab: consecutive tids -> consecutive pixels (coalesced 128B reads)
  for (int idx = tid; idx < HH * WW * CHG; idx += 256) {
    int ch  = idx >> 12;          // 0..CHG-1
    int pix = idx & (HH * WW - 1);
#if USE_ASYNC_LDS
    __builtin_amdgcn_global_load_async_to_lds_b32(
        (as1_int*)(img + (long long)ch * (HH * WW) + pix),
        (as3_int*)&tile[pix * CHG + ch], 0, 0);
#else
    tile[pix * CHG + ch] = img[(long long)ch * (HH * WW) + pix];
#endif
  }
  wait_async_zero();
  __syncthreads();

  const float* xys = out + XYS_OFF + (long long)bv * RR * SS * 2;
  const float* ovl = out + OVL_OFF + (long long)bv * RR;
  float* feat = out + (long long)bv * RR * SS * CCH;

  int cc = tid & (CHG - 1);
  int pb = tid >> 2;            // 0..63 : point slot within iteration

  for (int it = 0; it < RPB * SS / 64; ++it) {
    int point = it * 64 + pb;
    int rl = point >> 5;        // local ray
    int ss = point & (SS - 1);  // sample index
    int ray = rt * RPB + rl;
    long long pidx = (long long)ray * SS + ss;

    float xs = xys[pidx * 2 + 0];
    float ysm = xys[pidx * 2 + 1];
    float ovf = ovl[ray];

    // grid = 2*xy-1 ; x = (grid+1)*W/2 - 0.5 = xy*W - 0.5
    float x = xs * (float)WW - 0.5f;
    float y = ysm * (float)HH - 0.5f;
    float x0 = floorf(x), y0 = floorf(y);
    float wx1 = x - x0, wy1 = y - y0;
    float wx0 = 1.0f - wx1, wy0 = 1.0f - wy1;

    float acc = 0.0f;
#pragma unroll
    for (int tap = 0; tap < 4; ++tap) {
      float xf = x0 + (float)(tap & 1);
      float yf = y0 + (float)(tap >> 1);
      float w = ((tap & 1) ? wx1 : wx0) * ((tap >> 1) ? wy1 : wy0);
      bool inb = (xf >= 0.0f) && (xf < (float)WW) && (yf >= 0.0f) && (yf < (float)HH);
      float xcl = fminf(fmaxf(xf, 0.0f), (float)(WW - 1));
      float ycl = fminf(fmaxf(yf, 0.0f), (float)(HH - 1));
      int xi = (int)xcl, yi = (int)ycl;
      float v = tile[(yi * WW + xi) * CHG + cc];
      acc += inb ? (w * v) : 0.0f;
    }
    feat[pidx * CCH + cg * CHG + cc] = acc * ovf;
  }
}

// ---------------------------------------------------------------------------
extern "C" void kernel_launch(void* const* d_in, const int* in_sizes, int n_in,
                              void* d_out, int out_size, void* d_ws, size_t ws_size,
                              hipStream_t stream) {
  const float* images = (const float*)d_in[0];
  const float* ex     = (const float*)d_in[1];
  const float* K      = (const float*)d_in[2];
  const float* nearp  = (const float*)d_in[3];
  const float* farp   = (const float*)d_in[4];
  float* out = (float*)d_out;

  int nrays = BB * VV * RR;
  ray_setup_kernel<<<(nrays + 255) / 256, 256, 0, stream>>>(ex, K, nearp, farp, out);

  int nblk = BB * VV * (RR / RPB) * (CCH / CHG); // 2048
  sampler_kernel<<<nblk, 256, 0, stream>>>(images, out);
}